// GAT_69406671503628
// MI455X (gfx1250) — compile-verified
//
#include <hip/hip_runtime.h>
#include <math.h>

// ---------------------------------------------------------------------------
// GAT 3-layer forward for gfx1250 (MI455X).
//   N=50000 nodes, E=800000 edges (+N self loops), H=2 heads.
//   L1: 64 -> 2x64 (concat, relu)  L2: 128 -> 2x64 (concat, relu)
//   L3: 128 -> 2x16 (mean heads) -> softmax(16)
//
// GEMMs: V_WMMA_F32_16X16X4_F32, one wave per 16-row stripe (A-frag reuse).
// Edge phase: per-call CSR build (count/scan/fill), then one wave per dst
// node gathers its in-edges: wave-max, fused exp-sum + weighted feature
// accumulation in registers, single normalize + bias + relu store.
// No float atomics anywhere -> pure streaming reads at HBM rate.
// ---------------------------------------------------------------------------

typedef float v2f __attribute__((ext_vector_type(2)));
typedef float v8f __attribute__((ext_vector_type(8)));

#define NN 50000
#define EE 800000
#define ETOT (EE + NN)

__device__ __forceinline__ float lrelu(float v) { return v > 0.f ? v : 0.2f * v; }

__device__ __forceinline__ void edge_sd(int e, const int* __restrict__ e0,
                                        const int* __restrict__ e1, int& s, int& d) {
  if (e < EE) { s = e0[e]; d = e1[e]; } else { s = e - EE; d = s; }
}

// ---- fp32 WMMA GEMM: Hout[N,M] = X[N,K] * W[K,M] --------------------------
// One wave computes a 16 x M row stripe: A fragment loaded once per k-step,
// reused across all M/16 column tiles (W stays L2-resident).
template<int K, int M>
__global__ __launch_bounds__(32) void gemm_wmma(const float* __restrict__ X,
                                                const float* __restrict__ W,
                                                float* __restrict__ Hout) {
  constexpr int TM = M / 16;
  const int tm   = blockIdx.x;           // row tile (N/16 of them)
  const int lane = threadIdx.x;          // wave32
  const int r    = lane & 15;
  const int kb   = (lane >> 4) << 1;     // K sub-offset: 0 (lanes 0-15) or 2

  const float* xrow  = X + (size_t)(tm * 16 + r) * K;   // A row = lane%16
  const float* wbase = W + r;                           // B col base = lane%16

  v8f zero = {};
  v8f acc[TM];
#pragma unroll
  for (int t = 0; t < TM; ++t) acc[t] = zero;

  for (int k0 = 0; k0 < K; k0 += 4) {
    // A 16x4 fp32 frag: a[v] = A[row][k0 + kb + v]
    const float2 av = *(const float2*)(xrow + k0 + kb);
    v2f a; a.x = av.x; a.y = av.y;
    const float* wk = wbase + (size_t)(k0 + kb) * M;
#pragma unroll
    for (int t = 0; t < TM; ++t) {
      // B 4x16 fp32 frag: b[v] = B[k0 + kb + v][tile col]
      v2f b;
      b.x = wk[t * 16];
      b.y = wk[t * 16 + M];
      acc[t] = __builtin_amdgcn_wmma_f32_16x16x4_f32(false, a, false, b,
                                                     (short)0, acc[t], false, false);
    }
  }
  // C/D 16x16 fp32: VGPR v, lanes 0-15 -> row v, lanes 16-31 -> row v+8
  const int rowBase = tm * 16 + ((lane >> 4) << 3);
#pragma unroll
  for (int t = 0; t < TM; ++t) {
    float* o = Hout + (size_t)rowBase * M + t * 16 + r;
#pragma unroll
    for (int v = 0; v < 8; ++v) o[(size_t)v * M] = acc[t][v];
  }
}

// ---- per (node, head) attention logits: al = <h[n,hd,:], a_vec[hd,:]> -----
__global__ void attn_logits(const float* __restrict__ Hf,
                            const float* __restrict__ a_src,
                            const float* __restrict__ a_dst,
                            float* __restrict__ al_s, float* __restrict__ al_d,
                            int F) {
  int idx = blockIdx.x * blockDim.x + threadIdx.x;   // n*2 + hd
  if (idx >= NN * 2) return;
  int n = idx >> 1, hd = idx & 1;
  const float* hp = Hf + (size_t)n * (2 * F) + hd * F;
  const float* as = a_src + hd * F;
  const float* ad = a_dst + hd * F;
  float s = 0.f, d = 0.f;
  for (int f = 0; f < F; ++f) { float hv = hp[f]; s += hv * as[f]; d += hv * ad[f]; }
  al_s[idx] = s; al_d[idx] = d;
}

// ---- CSR build: count -> scan -> fill -------------------------------------
__global__ void csr_count(const int* __restrict__ e0, const int* __restrict__ e1,
                          int* __restrict__ deg) {
  int e = blockIdx.x * blockDim.x + threadIdx.x;
  if (e >= ETOT) return;
  int s, d; edge_sd(e, e0, e1, s, d);
  atomicAdd(&deg[d], 1);
}

__global__ __launch_bounds__(1024) void scan_deg(const int* __restrict__ deg,
                                                 int* __restrict__ off) {
  __shared__ int tmp[1024];
  __shared__ int carry_s;
  if (threadIdx.x == 0) carry_s = 0;
  __syncthreads();
  for (int base = 0; base < NN; base += 1024) {
    int i = base + (int)threadIdx.x;
    int v = (i < NN) ? deg[i] : 0;
    tmp[threadIdx.x] = v;
    __syncthreads();
#pragma unroll
    for (int o = 1; o < 1024; o <<= 1) {
      int t = (threadIdx.x >= (unsigned)o) ? tmp[threadIdx.x - o] : 0;
      __syncthreads();
      tmp[threadIdx.x] += t;
      __syncthreads();
    }
    int incl  = tmp[threadIdx.x];
    int carry = carry_s;
    if (i < NN) off[i] = carry + incl - v;   // exclusive scan
    __syncthreads();
    if (threadIdx.x == 1023) carry_s = carry + tmp[1023];
    __syncthreads();
  }
  if (threadIdx.x == 0) off[NN] = carry_s;
}

__global__ void csr_fill(const int* __restrict__ e0, const int* __restrict__ e1,
                         const int* __restrict__ off, int* __restrict__ cur,
                         int* __restrict__ col) {
  int e = blockIdx.x * blockDim.x + threadIdx.x;
  if (e >= ETOT) return;
  int s, d; edge_sd(e, e0, e1, s, d);
  int p = atomicAdd(&cur[d], 1);
  col[off[d] + p] = s;
}

// ---- per-node gather aggregation: segment softmax + weighted sum ---------
// One wave per destination node. F = per-head width, HF = 2*F.
template<int F, int HF>
__global__ __launch_bounds__(256) void gat_aggregate(
    const int* __restrict__ row_off, const int* __restrict__ col_src,
    const float* __restrict__ al_s, const float* __restrict__ al_d,
    const float* __restrict__ Hf, const float* __restrict__ bias,
    float* __restrict__ Out, int do_relu) {
  constexpr int V = HF / 32;                   // floats per lane (4 or 1)
  int node = (blockIdx.x * blockDim.x + threadIdx.x) >> 5;
  int lane = threadIdx.x & 31;
  if (node >= NN) return;
  int beg = row_off[node], end = row_off[node + 1];
  float ald0 = al_d[node * 2 + 0], ald1 = al_d[node * 2 + 1];

  // pass 1: segment max per head (wave-parallel over in-edges)
  float m0 = -INFINITY, m1 = -INFINITY;
  for (int i = beg + lane; i < end; i += 32) {
    int s = col_src[i];
    m0 = fmaxf(m0, lrelu(al_s[s * 2 + 0] + ald0));
    m1 = fmaxf(m1, lrelu(al_s[s * 2 + 1] + ald1));
  }
#pragma unroll
  for (int o = 16; o > 0; o >>= 1) {
    m0 = fmaxf(m0, __shfl_xor(m0, o));
    m1 = fmaxf(m1, __shfl_xor(m1, o));
  }

  // pass 2: fused exp-sum + weighted feature accumulation (registers only)
  float den0 = 0.f, den1 = 0.f;
  float acc[V];
#pragma unroll
  for (int v = 0; v < V; ++v) acc[v] = 0.f;

  for (int base = beg; base < end; base += 32) {
    int i = base + lane;
    int s = 0; float ex0 = 0.f, ex1 = 0.f;
    if (i < end) {
      s   = col_src[i];
      ex0 = expf(lrelu(al_s[s * 2 + 0] + ald0) - m0);
      ex1 = expf(lrelu(al_s[s * 2 + 1] + ald1) - m1);
      den0 += ex0; den1 += ex1;
    }
    int cnt = min(32, end - base);
    for (int j = 0; j < cnt; ++j) {
      int   sj = __shfl(s,   j);
      float e0 = __shfl(ex0, j);
      float e1 = __shfl(ex1, j);
      float myex = (lane < 16) ? e0 : e1;      // lane's features are one head
      if constexpr (V == 4) {
        const float4 hv = *(const float4*)(Hf + (size_t)sj * HF + lane * 4);
        acc[0] += myex * hv.x; acc[1] += myex * hv.y;
        acc[2] += myex * hv.z; acc[3] += myex * hv.w;
      } else {
        acc[0] += myex * Hf[(size_t)sj * HF + lane];
      }
    }
  }
#pragma unroll
  for (int o = 16; o > 0; o >>= 1) {
    den0 += __shfl_xor(den0, o);
    den1 += __shfl_xor(den1, o);
  }
  float inv = 1.f / (((lane < 16) ? den0 : den1) + 1e-16f);

  if constexpr (V == 4) {
    int f = lane * 4;
    float4 o4;
    o4.x = acc[0] * inv; o4.y = acc[1] * inv;
    o4.z = acc[2] * inv; o4.w = acc[3] * inv;
    if (bias) { o4.x += bias[f]; o4.y += bias[f + 1]; o4.z += bias[f + 2]; o4.w += bias[f + 3]; }
    if (do_relu) {
      o4.x = fmaxf(o4.x, 0.f); o4.y = fmaxf(o4.y, 0.f);
      o4.z = fmaxf(o4.z, 0.f); o4.w = fmaxf(o4.w, 0.f);
    }
    *(float4*)(Out + (size_t)node * HF + f) = o4;
  } else {
    float o1 = acc[0] * inv;
    if (bias) o1 += bias[lane];
    if (do_relu) o1 = fmaxf(o1, 0.f);
    Out[(size_t)node * HF + lane] = o1;
  }
}

// ---- layer 3: mean over heads + bias + row softmax -> d_out --------------
__global__ void finalize_k(const float* __restrict__ acc, const float* __restrict__ b,
                           float* __restrict__ out) {
  int n = blockIdx.x * blockDim.x + threadIdx.x;
  if (n >= NN) return;
  float v[16]; float mx = -INFINITY;
#pragma unroll
  for (int f = 0; f < 16; ++f) {
    v[f] = 0.5f * (acc[(size_t)n * 32 + f] + acc[(size_t)n * 32 + 16 + f]) + b[f];
    mx = fmaxf(mx, v[f]);
  }
  float sum = 0.f;
#pragma unroll
  for (int f = 0; f < 16; ++f) { v[f] = expf(v[f] - mx); sum += v[f]; }
  float inv = 1.f / sum;
#pragma unroll
  for (int f = 0; f < 16; ++f) out[(size_t)n * 16 + f] = v[f] * inv;
}

extern "C" void kernel_launch(void* const* d_in, const int* in_sizes, int n_in,
                              void* d_out, int out_size, void* d_ws, size_t ws_size,
                              hipStream_t stream) {
  const float* x0  = (const float*)d_in[0];
  const int*   ei  = (const int*)d_in[1];
  const int*   e0  = ei;          // src row
  const int*   e1  = ei + EE;     // dst row
  const float* W1  = (const float*)d_in[2];
  const float* a1s = (const float*)d_in[3];
  const float* a1d = (const float*)d_in[4];
  const float* b1  = (const float*)d_in[5];
  const float* W2  = (const float*)d_in[6];
  const float* a2s = (const float*)d_in[7];
  const float* a2d = (const float*)d_in[8];
  const float* b2  = (const float*)d_in[9];
  const float* W3  = (const float*)d_in[10];
  const float* a3s = (const float*)d_in[11];
  const float* a3d = (const float*)d_in[12];
  const float* b3  = (const float*)d_in[13];

  // workspace: h_buf[N*128] | xbuf[N*128] | al_s,al_d[N*2] | deg[N] | off[N+1] | col[ETOT]
  float* h_buf = (float*)d_ws;
  float* xbuf  = h_buf + (size_t)NN * 128;
  float* al_s  = xbuf  + (size_t)NN * 128;
  float* al_d  = al_s  + NN * 2;
  int*   deg   = (int*)(al_d + NN * 2);
  int*   off   = deg + NN;
  int*   col   = off + (NN + 1);

  const int gEdge = (ETOT + 255) / 256;
  const int gNH   = (NN * 2 + 255) / 256;
  const int gAgg  = (NN * 32 + 255) / 256;

  // ---------------- CSR build (shared by all 3 layers) --------------------
  hipMemsetAsync(deg, 0, NN * sizeof(int), stream);
  csr_count<<<gEdge, 256, 0, stream>>>(e0, e1, deg);
  scan_deg<<<1, 1024, 0, stream>>>(deg, off);
  hipMemsetAsync(deg, 0, NN * sizeof(int), stream);   // reuse as fill cursor
  csr_fill<<<gEdge, 256, 0, stream>>>(e0, e1, off, deg, col);

  // ---------------- Layer 1: x0[N,64] @ W1[64,128] ------------------------
  gemm_wmma<64, 128><<<NN / 16, 32, 0, stream>>>(x0, W1, h_buf);
  attn_logits<<<gNH, 256, 0, stream>>>(h_buf, a1s, a1d, al_s, al_d, 64);
  gat_aggregate<64, 128><<<gAgg, 256, 0, stream>>>(off, col, al_s, al_d,
                                                   h_buf, b1, xbuf, 1);

  // ---------------- Layer 2: xbuf[N,128] @ W2[128,128] --------------------
  gemm_wmma<128, 128><<<NN / 16, 32, 0, stream>>>(xbuf, W2, h_buf);
  attn_logits<<<gNH, 256, 0, stream>>>(h_buf, a2s, a2d, al_s, al_d, 64);
  gat_aggregate<64, 128><<<gAgg, 256, 0, stream>>>(off, col, al_s, al_d,
                                                   h_buf, b2, xbuf, 1);

  // ---------------- Layer 3: xbuf[N,128] @ W3[128,32] ---------------------
  gemm_wmma<128, 32><<<NN / 16, 32, 0, stream>>>(xbuf, W3, h_buf);
  attn_logits<<<gNH, 256, 0, stream>>>(h_buf, a3s, a3d, al_s, al_d, 16);
  gat_aggregate<16, 32><<<gAgg, 256, 0, stream>>>(off, col, al_s, al_d,
                                                  h_buf, (const float*)nullptr,
                                                  xbuf, 0);   // raw [N,32]
  finalize_k<<<(NN + 255) / 256, 256, 0, stream>>>(xbuf, b3, (float*)d_out);
}